// GCNGraphRegression_42159398977692
// MI455X (gfx1250) — compile-verified
//
#include <hip/hip_runtime.h>

#define NN 50000      // nodes  (multiple of 16)
#define NE 800000     // edges
#define HD 64         // hidden dim
#define NG 512        // graphs (multiple of 16)

typedef __attribute__((ext_vector_type(2))) float v2f;
typedef __attribute__((ext_vector_type(8))) float v8f;

__device__ __forceinline__ void atomAddF(float* p, float v) {
    unsafeAtomicAdd(p, v);   // lowers to global_atomic_add_f32 on gfx1250
}

// ---------------- degree / dinv ----------------
__global__ void k_deg_init(float* __restrict__ deg) {
    int i = blockIdx.x * blockDim.x + threadIdx.x;
    if (i < NN) deg[i] = 1.0f;               // self loop
}

__global__ void k_deg_count(const int* __restrict__ dst, float* __restrict__ deg) {
    int e = blockIdx.x * blockDim.x + threadIdx.x;
    if (e < NE) atomAddF(&deg[dst[e]], 1.0f);
}

__global__ void k_dinv(float* __restrict__ deg) {
    int i = blockIdx.x * blockDim.x + threadIdx.x;
    if (i < NN) deg[i] = rsqrtf(deg[i]);     // deg >= 1 always
}

// ---------------- WMMA matmul: Out = X(nrows x 64) @ W(64 x 64) ----------------
// Requires nrows % 16 == 0 (NN=50000 and NG=512 both are).
// MODE 0: store raw H, and Out = bias + H*dinv^2 (GCN self-loop init, pre-atomics)
// MODE 1: Out = relu(H + bias)   (lin1)
template<int MODE>
__global__ __launch_bounds__(256)
void k_wmma_mm(const float* __restrict__ X, const float* __restrict__ W,
               const float* __restrict__ bias, const float* __restrict__ dinv,
               float* __restrict__ H, float* __restrict__ Out, int nrows) {
    // W staged in LDS in B-fragment order: Wl[kc][ct][lane][pair]
    __shared__ float Wl[16][4][32][2];       // 16 KB
    for (int idx = threadIdx.x; idx < 4096; idx += 256) {
        const int p  = idx & 1;
        const int ln = (idx >> 1) & 31;
        const int ct = (idx >> 6) & 3;
        const int kc = idx >> 8;
        const int k  = kc * 4 + (ln >> 4) * 2 + p;
        const int n  = ct * 16 + (ln & 15);
        Wl[kc][ct][ln][p] = W[(size_t)k * HD + n];
    }
    __syncthreads();

    const int wave = (int)((blockIdx.x * blockDim.x + threadIdx.x) >> 5);
    const int lane = (int)(threadIdx.x & 31);
    const int r0 = wave * 16;
    if (r0 >= nrows) return;                 // wave-uniform exit (EXEC all-1 for WMMA)

    const int half = lane >> 4;              // 0: lanes 0-15, 1: lanes 16-31
    const int lr   = lane & 15;
    const int m    = r0 + lr;                // A rows: M = lane&15; half selects K pair

    v8f acc0 = {}, acc1 = {}, acc2 = {}, acc3 = {};

    for (int kc = 0; kc < 16; ++kc) {
        const int k0 = kc * 4 + half * 2;
        const v2f a = *(const v2f*)(X + (size_t)m * HD + k0);

        #pragma unroll
        for (int ct = 0; ct < 4; ++ct) {
            const v2f b = *(const v2f*)&Wl[kc][ct][lane][0];   // ds_load_b64
            v8f c = (ct == 0) ? acc0 : (ct == 1) ? acc1 : (ct == 2) ? acc2 : acc3;
            c = __builtin_amdgcn_wmma_f32_16x16x4_f32(false, a, false, b,
                                                      (short)0, c, false, false);
            if      (ct == 0) acc0 = c;
            else if (ct == 1) acc1 = c;
            else if (ct == 2) acc2 = c;
            else              acc3 = c;
        }
    }

    // C/D layout: VGPR j -> row M=j (lanes 0-15) / M=j+8 (lanes 16-31), N = lane&15
    const int rbase = r0 + half * 8;
    #pragma unroll
    for (int ct = 0; ct < 4; ++ct) {
        const int col = ct * 16 + lr;
        const v8f c = (ct == 0) ? acc0 : (ct == 1) ? acc1 : (ct == 2) ? acc2 : acc3;
        #pragma unroll
        for (int j = 0; j < 8; ++j) {
            const int row = rbase + j;
            const float v = c[j];
            if (MODE == 0) {
                H[(size_t)row * HD + col] = v;
                const float di = dinv[row];
                Out[(size_t)row * HD + col] = bias[col] + v * di * di;
            } else {
                Out[(size_t)row * HD + col] = fmaxf(v + bias[col], 0.0f);
            }
        }
    }
}

// ---------------- edge aggregation: Out[dst] += H[src] * dinv[src]*dinv[dst] ----
// 16 threads per edge, float4 per thread.
__global__ void k_agg_edges(const float* __restrict__ H, const int* __restrict__ src,
                            const int* __restrict__ dst, const float* __restrict__ dinv,
                            float* __restrict__ Out) {
    int t = blockIdx.x * blockDim.x + threadIdx.x;
    int e  = t >> 4;
    int h4 = (t & 15) * 4;
    if (e >= NE) return;
    int s = src[e], d = dst[e];
    float norm = dinv[s] * dinv[d];
    const float4 hv = *(const float4*)(H + (size_t)s * HD + h4);
    float* o = Out + (size_t)d * HD + h4;
    atomAddF(o + 0, hv.x * norm);
    atomAddF(o + 1, hv.y * norm);
    atomAddF(o + 2, hv.z * norm);
    atomAddF(o + 3, hv.w * norm);
}

__global__ void k_relu4(float4* __restrict__ x, int n4) {
    int i = blockIdx.x * blockDim.x + threadIdx.x;
    if (i < n4) {
        float4 v = x[i];
        v.x = fmaxf(v.x, 0.0f); v.y = fmaxf(v.y, 0.0f);
        v.z = fmaxf(v.z, 0.0f); v.w = fmaxf(v.w, 0.0f);
        x[i] = v;
    }
}

__global__ void k_zero(float* __restrict__ x, int n) {
    int i = blockIdx.x * blockDim.x + threadIdx.x;
    if (i < n) x[i] = 0.0f;
}

// ---------------- mean pool (16 threads per node, float4) ----------------
__global__ void k_pool(const float* __restrict__ X, const int* __restrict__ batch,
                       float* __restrict__ sums, float* __restrict__ cnt) {
    int t = blockIdx.x * blockDim.x + threadIdx.x;
    int node = t >> 4;
    int h4   = (t & 15) * 4;
    if (node >= NN) return;
    int g = batch[node];
    const float4 xv = *(const float4*)(X + (size_t)node * HD + h4);
    float* s = sums + (size_t)g * HD + h4;
    atomAddF(s + 0, xv.x);
    atomAddF(s + 1, xv.y);
    atomAddF(s + 2, xv.z);
    atomAddF(s + 3, xv.w);
    if ((t & 15) == 0) atomAddF(&cnt[g], 1.0f);
}

__global__ void k_pooled(const float* __restrict__ sums, const float* __restrict__ cnt,
                         float* __restrict__ pooled) {
    int t = blockIdx.x * blockDim.x + threadIdx.x;
    if (t >= NG * HD) return;
    int g = t >> 6;
    pooled[t] = sums[t] / fmaxf(cnt[g], 1.0f);
}

// ---------------- lin2: out[g] = dot(h1[g,:], w) + b ----------------
__global__ void k_lin2(const float* __restrict__ h1, const float* __restrict__ w,
                       const float* __restrict__ b, float* __restrict__ out) {
    int g = blockIdx.x * blockDim.x + threadIdx.x;
    if (g >= NG) return;
    float acc = b[0];
    #pragma unroll 8
    for (int k = 0; k < HD; ++k) acc += h1[(size_t)g * HD + k] * w[k];
    out[g] = acc;
}

static inline size_t alignup(size_t v) { return (v + 255) & ~(size_t)255; }

extern "C" void kernel_launch(void* const* d_in, const int* in_sizes, int n_in,
                              void* d_out, int out_size, void* d_ws, size_t ws_size,
                              hipStream_t stream) {
    const float* x     = (const float*)d_in[0];
    const int*   ei    = (const int*)  d_in[1];
    const int*   batch = (const int*)  d_in[2];
    const float* W0    = (const float*)d_in[3];
    const float* b0    = (const float*)d_in[4];
    const float* W1    = (const float*)d_in[5];
    const float* b1    = (const float*)d_in[6];
    const float* W2    = (const float*)d_in[7];
    const float* b2    = (const float*)d_in[8];
    const float* l1w   = (const float*)d_in[9];
    const float* l1b   = (const float*)d_in[10];
    const float* l2w   = (const float*)d_in[11];
    const float* l2b   = (const float*)d_in[12];
    float* out = (float*)d_out;

    const int* src = ei;
    const int* dst = ei + NE;

    char* ws = (char*)d_ws;
    size_t off = 0;
    float* dinv   = (float*)(ws + off); off += alignup((size_t)NN * 4);
    float* bufA   = (float*)(ws + off); off += alignup((size_t)NN * HD * 4);  // H = X@W
    float* bufB   = (float*)(ws + off); off += alignup((size_t)NN * HD * 4);  // agg / next X
    float* sums   = (float*)(ws + off); off += alignup((size_t)NG * HD * 4);
    float* cnt    = (float*)(ws + off); off += alignup((size_t)NG * 4);
    float* pooled = (float*)(ws + off); off += alignup((size_t)NG * HD * 4);
    float* h1     = (float*)(ws + off); off += alignup((size_t)NG * HD * 4);

    const int TB = 256;
    // degrees (shared across layers)
    k_deg_init <<<(NN + TB - 1) / TB, TB, 0, stream>>>(dinv);
    k_deg_count<<<(NE + TB - 1) / TB, TB, 0, stream>>>(dst, dinv);
    k_dinv     <<<(NN + TB - 1) / TB, TB, 0, stream>>>(dinv);

    const float* Ws[3] = {W0, W1, W2};
    const float* bs[3] = {b0, b1, b2};
    const float* Xcur = x;

    const int mmBlocksN = (((NN + 15) / 16) + 7) / 8;            // 8 waves/block
    const int aggBlocks = (int)(((size_t)NE * 16 + TB - 1) / TB);
    const int n4        = NN * HD / 4;
    const int r4Blocks  = (n4 + TB - 1) / TB;
    const int poolBlks  = (int)(((size_t)NN * 16 + TB - 1) / TB);

    for (int l = 0; l < 3; ++l) {
        // bufA <- Xcur @ W ; bufB <- bias + bufA*dinv^2 (self-loop term)
        k_wmma_mm<0><<<mmBlocksN, TB, 0, stream>>>(Xcur, Ws[l], bs[l], dinv, bufA, bufB, NN);
        // bufB[dst] += bufA[src] * dinv[src]*dinv[dst]
        k_agg_edges<<<aggBlocks, TB, 0, stream>>>(bufA, src, dst, dinv, bufB);
        // relu in place -> next layer's X
        k_relu4<<<r4Blocks, TB, 0, stream>>>((float4*)bufB, n4);
        Xcur = bufB;
    }

    // mean pool
    k_zero<<<((NG * HD) + TB - 1) / TB, TB, 0, stream>>>(sums, NG * HD);
    k_zero<<<(NG + TB - 1) / TB, TB, 0, stream>>>(cnt, NG);
    k_pool<<<poolBlks, TB, 0, stream>>>(Xcur, batch, sums, cnt);
    k_pooled<<<((NG * HD) + TB - 1) / TB, TB, 0, stream>>>(sums, cnt, pooled);

    // lin1 (WMMA, bias+relu fused) then lin2
    const int mmBlocksG = (((NG + 15) / 16) + 7) / 8;
    k_wmma_mm<1><<<mmBlocksG, TB, 0, stream>>>(pooled, l1w, l1b, nullptr, nullptr, h1, NG);
    k_lin2<<<(NG + TB - 1) / TB, TB, 0, stream>>>(h1, l2w, l2b, out);
}